// MedicalCrossModalLoss_45226005627391
// MI455X (gfx1250) — compile-verified
//
#include <hip/hip_runtime.h>
#include <math.h>

#define TAU   0.07f
#define ALPHA 0.3f
#define NB    8192      // batch
#define DD    256       // embedding dim
#define ROIHW 1024      // 32*32

typedef __attribute__((ext_vector_type(2))) float v2f;
typedef __attribute__((ext_vector_type(8))) float v8f;

// unrolled descending top-5 insertion (strict >, earlier entries win ties)
__device__ __forceinline__ void top5_insert(
    float v, int j,
    float& t0, float& t1, float& t2, float& t3, float& t4,
    int& c0, int& c1, int& c2, int& c3, int& c4)
{
    if (v > t4) {
        if (v > t0)      { t4=t3;c4=c3; t3=t2;c3=c2; t2=t1;c2=c1; t1=t0;c1=c0; t0=v;c0=j; }
        else if (v > t1) { t4=t3;c4=c3; t3=t2;c3=c2; t2=t1;c2=c1; t1=v;c1=j; }
        else if (v > t2) { t4=t3;c4=c3; t3=t2;c3=c2; t2=v;c2=j; }
        else if (v > t3) { t4=t3;c4=c3; t3=v;c3=j; }
        else             { t4=v;c4=j; }
    }
}

// ---------------------------------------------------------------------------
// Kernel 1: per-row prep. One block (256 threads) per row.
// ---------------------------------------------------------------------------
__global__ __launch_bounds__(256) void prep_kernel(
    const float* __restrict__ fa, const float* __restrict__ mri,
    const float* __restrict__ roi,
    float* __restrict__ fa_n, float* __restrict__ mri_n,
    float* __restrict__ roi_sim, float* __restrict__ denom_t2i)
{
    __shared__ float sf2[256], sm2[256], sfm[256], sw[256];
    __shared__ float inv_f, inv_m;
    const int i = blockIdx.x;
    const int t = threadIdx.x;

    float f = fa[(size_t)i * DD + t];
    float m = mri[(size_t)i * DD + t];
    sf2[t] = f * f;
    sm2[t] = m * m;
    sfm[t] = f * m;
    float w = 0.f;
#pragma unroll
    for (int k = 0; k < 4; ++k) w += roi[(size_t)i * ROIHW + t * 4 + k];
    sw[t] = w;
    __syncthreads();
    for (int off = 128; off > 0; off >>= 1) {
        if (t < off) {
            sf2[t] += sf2[t + off];
            sm2[t] += sm2[t + off];
            sfm[t] += sfm[t + off];
            sw[t]  += sw[t + off];
        }
        __syncthreads();
    }
    if (t == 0) {
        float nf = sqrtf(sf2[0]);
        float nm = sqrtf(sm2[0]);
        inv_f = 1.0f / fmaxf(nf, 1e-12f);
        inv_m = 1.0f / fmaxf(nm, 1e-12f);
        float W   = sw[0];
        float num = W * W * sfm[0];
        float den = fmaxf(W * W * nf * nm, 1e-8f);
        roi_sim[i]   = num / den;
        denom_t2i[i] = 0.0f;            // re-zero atomic target every launch
    }
    __syncthreads();
    fa_n[(size_t)i * DD + t]  = f * inv_f;
    mri_n[(size_t)i * DD + t] = m * inv_m;
}

// ---------------------------------------------------------------------------
// Kernel 2: fused sim-GEMM (fp32 WMMA) + per-row top-5 mining.
// 128 blocks x 256 threads (8 waves). Block owns a 64-row strip of sim,
// sweeps 8192 columns in 128-wide panels. Wave w computes 4 WMMA tiles:
// row subtile (w&3), col subtiles (w>>2)*4.. +3. Panels land in a
// double-buffered LDS staging area (row stride 129 -> conflict-free scans;
// one barrier per panel). ALL 256 threads mine top-5: 4 threads per row,
// each owning a 32-column quarter; four sorted 5-lists merged at the end.
// ---------------------------------------------------------------------------
__global__ __launch_bounds__(256) void gemm_topk_kernel(
    const float* __restrict__ fa_n, const float* __restrict__ mri_n,
    float* __restrict__ diag_sim, float* __restrict__ denom_i2t,
    float* __restrict__ denom_t2i)
{
    __shared__ float simLDS[2][64 * 129];   // 66 KB double-buffered panel
    __shared__ float mv[256 * 5];           // per-thread top-5 values
    __shared__ int   mc[256 * 5];           // per-thread top-5 columns
    __shared__ float diagLDS[64];

    const int tid  = threadIdx.x;
    const int lane = tid & 31;              // wave32
    const int wave = tid >> 5;
    const int row0 = blockIdx.x * 64;
    const int rsub = wave & 3;              // row subtile 0..3
    const int cgrp = wave >> 2;             // col group 0..1 (64 cols each)
    const int l15  = lane & 15;
    const int khalf = lane >> 4;            // f32 frag: lanes16-31 hold K=2,3

    // scan assignment: 4 threads per row, one 32-col quarter each
    const int srow = tid & 63;
    const int sq   = tid >> 6;              // quarter 0..3

    const float* aBase = fa_n + (size_t)(row0 + rsub * 16 + l15) * DD + khalf * 2;

    float tv0 = -INFINITY, tv1 = -INFINITY, tv2 = -INFINITY,
          tv3 = -INFINITY, tv4 = -INFINITY;
    int   tc0 = 0, tc1 = 0, tc2 = 0, tc3 = 0, tc4 = 0;
    float diagv  = 0.0f;
    int   hasDiag = 0;

    for (int n0 = 0; n0 < NB; n0 += 128) {
        if (n0 + 128 < NB) {                // prefetch next 128x256 B panel
            const float* pf = mri_n + (size_t)(n0 + 128) * DD;
            __builtin_prefetch(pf + (size_t)tid * 128, 0, 1);
        }

        v8f acc0 = {}, acc1 = {}, acc2 = {}, acc3 = {};
        const float* bBase =
            mri_n + (size_t)(n0 + cgrp * 64 + l15) * DD + khalf * 2;

#pragma unroll 4
        for (int k0 = 0; k0 < DD; k0 += 4) {
            v2f a  = *(const v2f*)(aBase + k0);
            v2f b0 = *(const v2f*)(bBase + k0);
            v2f b1 = *(const v2f*)(bBase + 16 * DD + k0);
            v2f b2 = *(const v2f*)(bBase + 32 * DD + k0);
            v2f b3 = *(const v2f*)(bBase + 48 * DD + k0);
            acc0 = __builtin_amdgcn_wmma_f32_16x16x4_f32(
                false, a, false, b0, (short)0, acc0, false, false);
            acc1 = __builtin_amdgcn_wmma_f32_16x16x4_f32(
                false, a, false, b1, (short)0, acc1, false, false);
            acc2 = __builtin_amdgcn_wmma_f32_16x16x4_f32(
                false, a, false, b2, (short)0, acc2, false, false);
            acc3 = __builtin_amdgcn_wmma_f32_16x16x4_f32(
                false, a, false, b3, (short)0, acc3, false, false);
        }

        const int buf = (n0 >> 7) & 1;
        // store fragments: VGPR p -> (M = p + 8*(lane>=16), N = lane&15)
        {
            const int mhi = khalf * 8;
#pragma unroll
            for (int p = 0; p < 8; ++p) {
                const int m = rsub * 16 + mhi + p;
                float* dst = &simLDS[buf][m * 129 + cgrp * 64 + l15];
                dst[0]  = acc0[p];
                dst[16] = acc1[p];
                dst[32] = acc2[p];
                dst[48] = acc3[p];
            }
        }
        // one barrier per panel: store(buf) visible to scanners; also
        // separates this store from the scan of buf two iterations ago.
        __syncthreads();

        {
            const int i = row0 + srow;
            const float* rowp = &simLDS[buf][srow * 129 + sq * 32];
            const int jbase   = n0 + sq * 32;
#pragma unroll 8
            for (int c = 0; c < 32; ++c) {
                const float v = rowp[c];
                const int   j = jbase + c;
                if (j == i) { diagv = v; hasDiag = 1; }
                else top5_insert(v, j, tv0,tv1,tv2,tv3,tv4, tc0,tc1,tc2,tc3,tc4);
            }
        }
    }

    // publish per-quarter lists + diagonal, then merge 4 lists per row
    mv[tid*5+0]=tv0; mv[tid*5+1]=tv1; mv[tid*5+2]=tv2; mv[tid*5+3]=tv3; mv[tid*5+4]=tv4;
    mc[tid*5+0]=tc0; mc[tid*5+1]=tc1; mc[tid*5+2]=tc2; mc[tid*5+3]=tc3; mc[tid*5+4]=tc4;
    if (hasDiag) diagLDS[srow] = diagv;
    __syncthreads();

    if (tid < 64) {
        const int i = row0 + tid;
        // start from quarter 0's sorted list
        float b0 = mv[tid*5+0], b1 = mv[tid*5+1], b2 = mv[tid*5+2],
              b3 = mv[tid*5+3], b4 = mv[tid*5+4];
        int   d0 = mc[tid*5+0], d1 = mc[tid*5+1], d2 = mc[tid*5+2],
              d3 = mc[tid*5+3], d4 = mc[tid*5+4];
#pragma unroll
        for (int q = 1; q < 4; ++q) {
            const int base = (q * 64 + tid) * 5;
#pragma unroll
            for (int e = 0; e < 5; ++e)
                top5_insert(mv[base + e], mc[base + e],
                            b0,b1,b2,b3,b4, d0,d1,d2,d3,d4);
        }
        const float e0 = expf(b0 / TAU);
        const float e1 = expf(b1 / TAU);
        const float e2 = expf(b2 / TAU);
        const float e3 = expf(b3 / TAU);
        const float e4 = expf(b4 / TAU);
        denom_i2t[i] = e0 + e1 + e2 + e3 + e4;
        diag_sim[i]  = diagLDS[tid];
        atomicAdd(&denom_t2i[d0], e0);
        atomicAdd(&denom_t2i[d1], e1);
        atomicAdd(&denom_t2i[d2], e2);
        atomicAdd(&denom_t2i[d3], e3);
        atomicAdd(&denom_t2i[d4], e4);
    }
}

// ---------------------------------------------------------------------------
// Kernel 3: finalize. One 1024-thread block.
//  global_loss = -0.5 * mean(2*diag/tau - log d1 - log d2)
//  out[i] = global_loss + 0.3 * (-log_softmax(roi_sim/tau))[i]
// ---------------------------------------------------------------------------
__global__ __launch_bounds__(1024) void finalize_kernel(
    const float* __restrict__ diag_sim, const float* __restrict__ denom_i2t,
    const float* __restrict__ denom_t2i, const float* __restrict__ roi_sim,
    float* __restrict__ out)
{
    __shared__ float rg[1024], rm[1024], rs[1024];
    const int t = threadIdx.x;

    float gsum = 0.f, m = -INFINITY, s = 0.f;
    for (int i = t; i < NB; i += 1024) {
        const float lp = diag_sim[i] / TAU;                  // log(pos)
        gsum += 2.0f * lp - logf(denom_i2t[i]) - logf(denom_t2i[i]);
        const float x = roi_sim[i] / TAU;                    // online logsumexp
        if (x > m) { s = s * expf(m - x) + 1.0f; m = x; }
        else       { s += expf(x - m); }
    }
    rg[t] = gsum; rm[t] = m; rs[t] = s;
    __syncthreads();
    for (int off = 512; off > 0; off >>= 1) {
        if (t < off) {
            rg[t] += rg[t + off];
            const float m2 = rm[t + off], s2 = rs[t + off];
            const float M  = fmaxf(rm[t], m2);
            rs[t] = rs[t] * expf(rm[t] - M) + s2 * expf(m2 - M);
            rm[t] = M;
        }
        __syncthreads();
    }
    const float gl   = -0.5f * rg[0] / (float)NB;
    const float xmax = rm[0];
    const float lse  = logf(rs[0]);
    for (int i = t; i < NB; i += 1024) {
        const float x = roi_sim[i] / TAU;
        out[i] = gl + ALPHA * (xmax + lse - x);   // -log_softmax
    }
}

// ---------------------------------------------------------------------------
extern "C" void kernel_launch(void* const* d_in, const int* in_sizes, int n_in,
                              void* d_out, int out_size, void* d_ws, size_t ws_size,
                              hipStream_t stream)
{
    (void)in_sizes; (void)n_in; (void)out_size; (void)ws_size;
    const float* fa  = (const float*)d_in[0];   // [8192,256]
    const float* mri = (const float*)d_in[1];   // [8192,256]
    const float* roi = (const float*)d_in[2];   // [8192,32,32]
    float* out = (float*)d_out;                 // [8192]

    // workspace layout (floats): ~16.9 MB
    float* ws        = (float*)d_ws;
    float* fa_n      = ws;                                   // 8192*256
    float* mri_n     = fa_n  + (size_t)NB * DD;              // 8192*256
    float* roi_sim   = mri_n + (size_t)NB * DD;              // 8192
    float* diag_sim  = roi_sim  + NB;                        // 8192
    float* denom_i2t = diag_sim + NB;                        // 8192
    float* denom_t2i = denom_i2t + NB;                       // 8192

    prep_kernel<<<NB, 256, 0, stream>>>(fa, mri, roi, fa_n, mri_n,
                                        roi_sim, denom_t2i);
    gemm_topk_kernel<<<NB / 64, 256, 0, stream>>>(fa_n, mri_n, diag_sim,
                                                  denom_i2t, denom_t2i);
    finalize_kernel<<<1, 1024, 0, stream>>>(diag_sim, denom_i2t, denom_t2i,
                                            roi_sim, out);
}